// StageNet_87316685128219
// MI455X (gfx1250) — compile-verified
//
#include <hip/hip_runtime.h>
#include <math.h>

// Problem constants
#define BB   256
#define VV   50
#define NN_  32
#define DD   128
#define HH   384
#define LL   12
#define CS_  10
#define OUT_ 128
#define ND_  4096
#define GG   1560
#define GP   1600          // G padded to multiple of 64
#define HS_  64
#define KH_  (HH*CS_)      // 3840
#define VH_  (VV*HH)       // 19200

typedef __attribute__((ext_vector_type(2))) float v2f;
typedef __attribute__((ext_vector_type(8))) float v8f;

__device__ __forceinline__ v8f wmma4(v2f a, v2f b, v8f c) {
  // D = A(16x4) * B(4x16) + C(16x16), all f32
  return __builtin_amdgcn_wmma_f32_16x16x4_f32(false, a, false, b, (short)0, c,
                                               false, false);
}

// Software-pipelined K-loop for a 16x64 tile (4 accumulators), all-global.
template <int K>
__device__ __forceinline__ void gemm_k4(const float* __restrict__ a,
                                        const float* __restrict__ b0,
                                        const float* __restrict__ b1,
                                        const float* __restrict__ b2,
                                        const float* __restrict__ b3,
                                        int klo, v8f& c0, v8f& c1, v8f& c2,
                                        v8f& c3) {
  v2f ac, f0, f1, f2, f3;
  ac.x = a[klo];  ac.y = a[klo + 1];
  f0.x = b0[klo]; f0.y = b0[klo + 1];
  f1.x = b1[klo]; f1.y = b1[klo + 1];
  f2.x = b2[klo]; f2.y = b2[klo + 1];
  f3.x = b3[klo]; f3.y = b3[klo + 1];
#pragma unroll 4
  for (int k0 = 0; k0 < K - 4; k0 += 4) {
    const int kn = k0 + 4 + klo;
    v2f an, g0, g1, g2, g3;
    an.x = a[kn];  an.y = a[kn + 1];
    g0.x = b0[kn]; g0.y = b0[kn + 1];
    g1.x = b1[kn]; g1.y = b1[kn + 1];
    g2.x = b2[kn]; g2.y = b2[kn + 1];
    g3.x = b3[kn]; g3.y = b3[kn + 1];
    c0 = wmma4(ac, f0, c0);
    c1 = wmma4(ac, f1, c1);
    c2 = wmma4(ac, f2, c2);
    c3 = wmma4(ac, f3, c3);
    ac = an; f0 = g0; f1 = g1; f2 = g2; f3 = g3;
  }
  c0 = wmma4(ac, f0, c0);
  c1 = wmma4(ac, f1, c1);
  c2 = wmma4(ac, f2, c2);
  c3 = wmma4(ac, f3, c3);
}

// ---------------------------------------------------------------------------
// zero-init scratch region (h, c, hwin, dwin are contiguous)
__global__ void k_zero(float* __restrict__ p, int n) {
  int i = blockIdx.x * blockDim.x + threadIdx.x;
  if (i < n) p[i] = 0.f;
}

// ---------------------------------------------------------------------------
// Fused gather + input GEMM, workgroup-cooperative with async LDS staging.
//   xo_in[v*BB + b][g] = sum_k embed[node_ids[b,v,k/128]][k%128] * kernel_w[g][k]
//                        + time[b,v]*kernel_w[g][ND] + kernel_b[g]
// Block (4 waves) computes a 64M x 64N tile; the shared B panel (64 rows of
// kernel_w) is staged into LDS in 32-wide K chunks, double-buffered via
// global_load_async_to_lds_b128 + s_wait_asynccnt + workgroup barriers.
// A (gathered embed rows) stays on the pipelined global path per wave; the
// node-id indirection is pipelined one full 128-K block ahead.
__global__ __launch_bounds__(128) void k_input_gemm(
    const int* __restrict__ node_ids, const float* __restrict__ time,
    const float* __restrict__ embed, const float* __restrict__ kw,
    const float* __restrict__ kb, float* __restrict__ xo_in) {
  const int w    = threadIdx.x >> 5;           // wave in block (0..3)
  const int lane = threadIdx.x & 31;
  const int gtg = blockIdx.x % (GP / 64);      // 25 N-tile groups
  const int mtb = blockIdx.x / (GP / 64);      // 200 M64 tiles
  const int m0b = mtb * 64;                    // block M base
  const int g0  = gtg * 64;                    // block N base (shared panel)
  const int m0  = m0b + w * 16;                // this wave's M base
  const int lr  = lane & 15;
  const int klo = (lane >> 4) * 2;

  const int rowm = m0 + lr;                    // A row (m = v*BB + b)
  const int ab = rowm & (BB - 1);
  const int av = rowm >> 8;
  const int* idsp = node_ids + (ab * VV + av) * NN_;

  // double-buffered B panel: 64 rows x 32 k floats = 8 KB per buffer
  __shared__ float bp[2][64 * 32];

  // stage one 32-wide K chunk of the B panel into bp[bufsel]
  auto stage = [&](int kc, int bufsel) {
    const int k0 = kc * 32;
#pragma unroll
    for (int i = 0; i < 4; ++i) {
      const int li = w * 128 + i * 32 + lane;  // 512 16-byte transfers total
      const int row = li >> 3;                 // panel row (0..63)
      const int seg = li & 7;                  // 16-byte segment in row
      int grow = g0 + row;
      if (grow >= GG) grow = GG - 1;           // clamp (stores are guarded)
      const float* gsrc = kw + (size_t)grow * (ND_ + 1) + k0 + seg * 4;
      const unsigned dst =
          (unsigned)(unsigned long long)&bp[bufsel][row * 32 + seg * 4];
      asm volatile("global_load_async_to_lds_b128 %0, %1, off"
                   :: "v"(dst), "v"(gsrc)
                   : "memory");
    }
  };

  v8f c0 = {}, c1 = {}, c2 = {}, c3 = {};

  stage(0, 0);
  int id_cur = idsp[0];
  int id_nxt = id_cur;
  const int NCH = ND_ / 32;                    // 128 chunks
  for (int kc = 0; kc < NCH; ++kc) {
    const int buf = kc & 1;
    if (kc + 1 < NCH) {
      stage(kc + 1, buf ^ 1);
      asm volatile("s_wait_asynccnt 0x4" ::: "memory");
    } else {
      asm volatile("s_wait_asynccnt 0x0" ::: "memory");
    }
    // prefetch next embed-row id one full 128-K block ahead
    if ((kc & 3) == 3 && (kc >> 2) + 1 < NN_) id_nxt = idsp[(kc >> 2) + 1];
    __syncthreads();                           // panel chunk visible to all

    const float* arow = embed + (size_t)id_cur * DD + (kc & 3) * 32;
    const float* bbase = &bp[buf][0];

    // 8 pipelined k-steps: A from global, B from LDS
    v2f ac, f0, f1, f2, f3;
    ac.x = arow[klo];                    ac.y = arow[klo + 1];
    f0.x = bbase[(0 * 16 + lr) * 32 + klo]; f0.y = bbase[(0 * 16 + lr) * 32 + klo + 1];
    f1.x = bbase[(1 * 16 + lr) * 32 + klo]; f1.y = bbase[(1 * 16 + lr) * 32 + klo + 1];
    f2.x = bbase[(2 * 16 + lr) * 32 + klo]; f2.y = bbase[(2 * 16 + lr) * 32 + klo + 1];
    f3.x = bbase[(3 * 16 + lr) * 32 + klo]; f3.y = bbase[(3 * 16 + lr) * 32 + klo + 1];
#pragma unroll
    for (int s = 0; s < 7; ++s) {
      const int kn = (s + 1) * 4 + klo;
      v2f an, g0f, g1f, g2f, g3f;
      an.x = arow[kn];                      an.y = arow[kn + 1];
      g0f.x = bbase[(0 * 16 + lr) * 32 + kn]; g0f.y = bbase[(0 * 16 + lr) * 32 + kn + 1];
      g1f.x = bbase[(1 * 16 + lr) * 32 + kn]; g1f.y = bbase[(1 * 16 + lr) * 32 + kn + 1];
      g2f.x = bbase[(2 * 16 + lr) * 32 + kn]; g2f.y = bbase[(2 * 16 + lr) * 32 + kn + 1];
      g3f.x = bbase[(3 * 16 + lr) * 32 + kn]; g3f.y = bbase[(3 * 16 + lr) * 32 + kn + 1];
      c0 = wmma4(ac, f0, c0);
      c1 = wmma4(ac, f1, c1);
      c2 = wmma4(ac, f2, c2);
      c3 = wmma4(ac, f3, c3);
      ac = an; f0 = g0f; f1 = g1f; f2 = g2f; f3 = g3f;
    }
    c0 = wmma4(ac, f0, c0);
    c1 = wmma4(ac, f1, c1);
    c2 = wmma4(ac, f2, c2);
    c3 = wmma4(ac, f3, c3);

    __syncthreads();                           // done reading before restage
    if ((kc & 3) == 3) id_cur = id_nxt;        // rotate id pipeline
  }

  const int hi = lane >> 4;
  v8f accv[4] = {c0, c1, c2, c3};
#pragma unroll
  for (int nt = 0; nt < 4; ++nt) {
    const int g = g0 + nt * 16 + lr;
    if (g >= GG) continue;
    const float kwlast = kw[(size_t)g * (ND_ + 1) + ND_];
    const float kbg = kb[g];
#pragma unroll
    for (int r = 0; r < 8; ++r) {
      const int m = m0 + r + hi * 8;
      const int b_ = m & (BB - 1), v_ = m >> 8;
      const float tv = time[b_ * VV + v_];
      xo_in[(size_t)m * GP + g] = accv[nt][r] + tv * kwlast + kbg;
    }
  }
}

// ---------------------------------------------------------------------------
// Per-step recurrent GEMM:
//   xo[b][g] = xo_in[t*BB + b][g] + sum_k h[b][k]*rec_w[g][k]
//              + time[b,t]*rec_w[g][H] + rec_b[g]
// M=256, N=GP, K=384.
__global__ __launch_bounds__(128) void k_rec_gemm(
    const float* __restrict__ h, const float* __restrict__ rw,
    const float* __restrict__ rb, const float* __restrict__ time,
    const float* __restrict__ xo_in, float* __restrict__ xo, int t) {
  const int wave = blockIdx.x * 4 + (threadIdx.x >> 5);
  const int lane = threadIdx.x & 31;
  const int ntg = wave % (GP / 64);
  const int mt  = wave / (GP / 64);          // 16 M tiles
  const int m0 = mt * 16, g0 = ntg * 64;
  const int lr = lane & 15;
  const int klo = (lane >> 4) * 2;

  const float* arow = h + (size_t)(m0 + lr) * HH;
  int g_cl[4];
#pragma unroll
  for (int nt = 0; nt < 4; ++nt) {
    int g = g0 + nt * 16 + lr;
    g_cl[nt] = (g >= GG) ? 0 : g;
  }
  v8f c0 = {}, c1 = {}, c2 = {}, c3 = {};
  gemm_k4<HH>(arow, rw + (size_t)g_cl[0] * (HH + 1),
              rw + (size_t)g_cl[1] * (HH + 1), rw + (size_t)g_cl[2] * (HH + 1),
              rw + (size_t)g_cl[3] * (HH + 1), klo, c0, c1, c2, c3);

  const int hi = lane >> 4;
  v8f accv[4] = {c0, c1, c2, c3};
#pragma unroll
  for (int nt = 0; nt < 4; ++nt) {
    const int g = g0 + nt * 16 + lr;
    if (g >= GG) continue;
    const float rwlast = rw[(size_t)g * (HH + 1) + HH];
    const float rbg = rb[g];
#pragma unroll
    for (int r = 0; r < 8; ++r) {
      const int m = m0 + r + hi * 8;
      const float tv = time[m * VV + t];
      xo[(size_t)m * GP + g] =
          accv[nt][r] + xo_in[((size_t)t * BB + m) * GP + g] + tv * rwlast + rbg;
    }
  }
}

// ---------------------------------------------------------------------------
// Per-step elementwise: cumax gates, cell update, window, ld softmax,
// lh materialization (lhA), theme MLP. One block per batch element b.
__global__ __launch_bounds__(384) void k_step(
    const float* __restrict__ xo, float* __restrict__ c, float* __restrict__ h,
    float* __restrict__ hwin, float* __restrict__ dwin,
    float* __restrict__ ldbuf, float* __restrict__ lhA,
    float* __restrict__ theme_g, const float* __restrict__ scale_w,
    const float* __restrict__ scale_b, const float* __restrict__ rescale_w,
    const float* __restrict__ rescale_b, float* __restrict__ dists, int t) {
  const int b = blockIdx.x;
  const int tid = threadIdx.x;
  const int slot = t % CS_;

  __shared__ float sh_fm[LL], sh_im[LL], sh_ld[CS_], sh_mlh[HH], sh_s[HS_];

  const float* xr = xo + (size_t)b * GP;

  if (tid == 0) {
    float xs[LL], p[LL];
    // fm = cumsum(softmax(xo[:, :L]))
    float mx = -1e30f;
    for (int i = 0; i < LL; ++i) { xs[i] = xr[i]; mx = fmaxf(mx, xs[i]); }
    float s = 0.f;
    for (int i = 0; i < LL; ++i) { p[i] = expf(xs[i] - mx); s += p[i]; }
    float cum = 0.f, fmsum = 0.f;
    for (int i = 0; i < LL; ++i) {
      cum += p[i];
      const float fv = cum / s;
      sh_fm[i] = fv;
      fmsum += fv;
    }
    const float d2 = 1.f - fmsum / (float)LL;   // dist = 1 - mean(fm)
    // im = suffix-sum of softmax(xo[:, L:2L])
    mx = -1e30f;
    for (int i = 0; i < LL; ++i) { xs[i] = xr[LL + i]; mx = fmaxf(mx, xs[i]); }
    s = 0.f;
    for (int i = 0; i < LL; ++i) { p[i] = expf(xs[i] - mx); s += p[i]; }
    cum = 0.f;
    for (int i = LL - 1; i >= 0; --i) { cum += p[i]; sh_im[i] = cum / s; }
    dwin[slot * BB + b] = d2;
    dists[t * BB + b] = d2;
    // ld = softmax(cumsum(chronological window of dist))
    float cv[CS_];
    float cc = 0.f;
    for (int k = 0; k < CS_; ++k) {
      const int tt = t - (CS_ - 1) + k;
      float dv = 0.f;
      if (tt == t) dv = d2;
      else if (tt >= 0) dv = dwin[(tt % CS_) * BB + b];
      cc += dv;
      cv[k] = cc;
    }
    mx = -1e30f;
    for (int k = 0; k < CS_; ++k) mx = fmaxf(mx, cv[k]);
    s = 0.f;
    for (int k = 0; k < CS_; ++k) { cv[k] = expf(cv[k] - mx); s += cv[k]; }
    for (int k = 0; k < CS_; ++k) {
      const float lv = cv[k] / s;
      sh_ld[k] = lv;
      ldbuf[b * CS_ + k] = lv;
    }
  }
  __syncthreads();

  // gates + cell update (one thread per hidden unit)
  {
    const int l = tid >> 5;          // chunk index (L=12 chunks of CH=32)
    const int ch = tid & 31;
    const float gf = xr[2 * LL + (0 * LL + l) * 32 + ch];
    const float gi = xr[2 * LL + (1 * LL + l) * 32 + ch];
    const float go = xr[2 * LL + (2 * LL + l) * 32 + ch];
    const float gc = xr[2 * LL + (3 * LL + l) * 32 + ch];
    const float f = 1.f / (1.f + expf(-gf));
    const float i_ = 1.f / (1.f + expf(-gi));
    const float o = 1.f / (1.f + expf(-go));
    const float cin = tanhf(gc);
    const float fm = sh_fm[l], im = sh_im[l], ov = fm * im;
    const float cl = c[b * HH + tid];
    const float cn = ov * (f * cl + i_ * cin) + (fm - ov) * cl + (im - ov) * cin;
    const float hn = o * tanhf(cn);
    c[b * HH + tid] = cn;
    h[b * HH + tid] = hn;
    hwin[slot * (BB * HH) + b * HH + tid] = hn;
  }
  __syncthreads();

  // lhA[b][h*CS + k] = ld[k] * hwin_chron[k][b][h];  mlh = mean_k
  {
    float acc = 0.f;
    const int base = b * KH_ + tid * CS_;
#pragma unroll
    for (int k = 0; k < CS_; ++k) {
      const int tt = t - (CS_ - 1) + k;
      float hv = 0.f;
      if (tt >= 0) hv = hwin[(tt % CS_) * (BB * HH) + b * HH + tid];
      const float lv = sh_ld[k] * hv;
      lhA[base + k] = lv;
      acc += lv;
    }
    sh_mlh[tid] = acc * (1.f / (float)CS_);
  }
  __syncthreads();

  // theme MLP: 384 -> 64 (relu) -> 384 (sigmoid)
  if (tid < HS_) {
    float acc = scale_b[tid];
    const float* wrow = scale_w + tid * HH;
    for (int hh = 0; hh < HH; ++hh) acc += sh_mlh[hh] * wrow[hh];
    sh_s[tid] = fmaxf(acc, 0.f);
  }
  __syncthreads();
  {
    float acc = rescale_b[tid];
    const float* wrow = rescale_w + tid * HS_;
#pragma unroll 8
    for (int j = 0; j < HS_; ++j) acc += sh_s[j] * wrow[j];
    theme_g[b * HH + tid] = 1.f / (1.f + expf(-acc));
  }
}

// ---------------------------------------------------------------------------
// Per-step conv einsum as GEMM:  conv[b][o] = sum_{kk<3840} lhA[b][kk]*conv_w[o][kk]
// then rnn[b][t][o] = theme[b][o]*(conv+conv_b[o]) + h[b][o]
__global__ __launch_bounds__(128) void k_conv(
    const float* __restrict__ lhA, const float* __restrict__ conv_w,
    const float* __restrict__ conv_b, const float* __restrict__ theme_g,
    const float* __restrict__ h, float* __restrict__ rnn, int t) {
  const int wave = blockIdx.x * 4 + (threadIdx.x >> 5);
  const int lane = threadIdx.x & 31;
  const int ntg = wave % (HH / 64);          // 6 groups
  const int mt  = wave / (HH / 64);          // 16 M tiles
  const int m0 = mt * 16, g0 = ntg * 64;
  const int lr = lane & 15;
  const int klo = (lane >> 4) * 2;

  const float* arow = lhA + (size_t)(m0 + lr) * KH_;
  v8f c0 = {}, c1 = {}, c2 = {}, c3 = {};
  gemm_k4<KH_>(arow, conv_w + (size_t)(g0 + 0 * 16 + lr) * KH_,
               conv_w + (size_t)(g0 + 1 * 16 + lr) * KH_,
               conv_w + (size_t)(g0 + 2 * 16 + lr) * KH_,
               conv_w + (size_t)(g0 + 3 * 16 + lr) * KH_, klo, c0, c1, c2, c3);

  const int hi = lane >> 4;
  v8f accv[4] = {c0, c1, c2, c3};
#pragma unroll
  for (int nt = 0; nt < 4; ++nt) {
    const int o = g0 + nt * 16 + lr;
    const float cb = conv_b[o];
#pragma unroll
    for (int r = 0; r < 8; ++r) {
      const int m = m0 + r + hi * 8;      // batch index b
      const float conv = accv[nt][r] + cb;
      const float rv = theme_g[m * HH + o] * conv + h[m * HH + o];
      rnn[(size_t)m * VH_ + t * HH + o] = rv;
    }
  }
}

// ---------------------------------------------------------------------------
// Final output GEMM: out[b][j] = sum_{p<19200} rnn[b][p]*out_w[j][p] + out_b[j]
__global__ __launch_bounds__(128) void k_out(
    const float* __restrict__ rnn, const float* __restrict__ ow,
    const float* __restrict__ ob, float* __restrict__ out) {
  const int wave = blockIdx.x * 4 + (threadIdx.x >> 5);
  const int lane = threadIdx.x & 31;
  const int ntg = wave % (OUT_ / 64);        // 2 groups
  const int mt  = wave / (OUT_ / 64);        // 16 M tiles
  const int m0 = mt * 16, g0 = ntg * 64;
  const int lr = lane & 15;
  const int klo = (lane >> 4) * 2;

  const float* arow = rnn + (size_t)(m0 + lr) * VH_;
  v8f c0 = {}, c1 = {}, c2 = {}, c3 = {};
  gemm_k4<VH_>(arow, ow + (size_t)(g0 + 0 * 16 + lr) * VH_,
               ow + (size_t)(g0 + 1 * 16 + lr) * VH_,
               ow + (size_t)(g0 + 2 * 16 + lr) * VH_,
               ow + (size_t)(g0 + 3 * 16 + lr) * VH_, klo, c0, c1, c2, c3);

  const int hi = lane >> 4;
  v8f accv[4] = {c0, c1, c2, c3};
#pragma unroll
  for (int nt = 0; nt < 4; ++nt) {
    const int j = g0 + nt * 16 + lr;
    const float bj = ob[j];
#pragma unroll
    for (int r = 0; r < 8; ++r) {
      const int m = m0 + r + hi * 8;
      out[m * OUT_ + j] = accv[nt][r] + bj;
    }
  }
}

// ---------------------------------------------------------------------------
extern "C" void kernel_launch(void* const* d_in, const int* in_sizes, int n_in,
                              void* d_out, int out_size, void* d_ws,
                              size_t ws_size, hipStream_t stream) {
  const int* node_ids   = (const int*)d_in[0];
  const float* time     = (const float*)d_in[3];
  const float* embed    = (const float*)d_in[6];
  const float* kernel_w = (const float*)d_in[7];
  const float* kernel_b = (const float*)d_in[8];
  const float* rec_w    = (const float*)d_in[9];
  const float* rec_b    = (const float*)d_in[10];
  const float* scale_w  = (const float*)d_in[11];
  const float* scale_b  = (const float*)d_in[12];
  const float* rescale_w= (const float*)d_in[13];
  const float* rescale_b= (const float*)d_in[14];
  const float* conv_w   = (const float*)d_in[15];
  const float* conv_b   = (const float*)d_in[16];
  const float* out_w    = (const float*)d_in[17];
  const float* out_b    = (const float*)d_in[18];

  float* out   = (float*)d_out;
  float* dists = out + BB * OUT_;

  float* ws = (float*)d_ws;
  const size_t off_xoin  = 0;
  const size_t off_xo    = off_xoin + (size_t)VV * BB * GP;
  const size_t off_h     = off_xo + (size_t)BB * GP;
  const size_t off_c     = off_h + (size_t)BB * HH;
  const size_t off_hwin  = off_c + (size_t)BB * HH;
  const size_t off_dwin  = off_hwin + (size_t)CS_ * BB * HH;
  const size_t off_ld    = off_dwin + (size_t)CS_ * BB;
  const size_t off_lhA   = off_ld + (size_t)BB * CS_;
  const size_t off_theme = off_lhA + (size_t)BB * KH_;
  const size_t off_rnn   = off_theme + (size_t)BB * HH;

  float* xo_in = ws + off_xoin;
  float* xo    = ws + off_xo;
  float* h     = ws + off_h;
  float* c     = ws + off_c;
  float* hwin  = ws + off_hwin;
  float* dwin  = ws + off_dwin;
  float* ldbuf = ws + off_ld;
  float* lhA   = ws + off_lhA;
  float* theme = ws + off_theme;
  float* rnn   = ws + off_rnn;

  // zero h, c, hwin, dwin (contiguous region)
  {
    const int nz = BB * HH * 2 + CS_ * BB * HH + CS_ * BB;
    k_zero<<<(nz + 255) / 256, 256, 0, stream>>>(h, nz);
  }

  // fused gather + input GEMM: 200 M64-tiles x 25 N-groups = 5000 blocks
  k_input_gemm<<<5000, 128, 0, stream>>>(node_ids, time, embed, kernel_w,
                                         kernel_b, xo_in);

  for (int t = 0; t < VV; ++t) {
    // recurrent GEMM: 16 x 25 = 400 waves
    k_rec_gemm<<<100, 128, 0, stream>>>(h, rec_w, rec_b, time, xo_in, xo, t);
    // per-batch elementwise
    k_step<<<BB, HH, 0, stream>>>(xo, c, h, hwin, dwin, ldbuf, lhA, theme,
                                  scale_w, scale_b, rescale_w, rescale_b,
                                  dists, t);
    // conv GEMM: 16 x 6 = 96 waves
    k_conv<<<24, 128, 0, stream>>>(lhA, conv_w, conv_b, theme, h, rnn, t);
  }

  // final output GEMM: 16 x 2 = 32 waves
  k_out<<<8, 128, 0, stream>>>(rnn, out_w, out_b, out);
}